// SelectiveSSM_43224550867174
// MI455X (gfx1250) — compile-verified
//
#include <hip/hip_runtime.h>
#include <hip/hip_bf16.h>

// ---------------------------------------------------------------------------
// Mamba selective-SSM forward on gfx1250 (MI455X), fp32 throughout.
// Matrix work uses V_WMMA_F32_16X16X4_F32 (wave32 matrix pipe), with GEMM
// shapes baked in at compile time so B-fragment addresses fold into the
// 24-bit immediate offsets of global_load and the K loop is one pointer bump.
// Each wave produces a 16x64 C strip (4 accumulators) to reuse the A fragment
// across 4 WMMA issues. Whole working set (~100 MB) fits the 192 MB L2.
// ---------------------------------------------------------------------------

#define BB        2
#define LL        1024
#define DMODEL    1024
#define DINNER    2048
#define NSTATE    16
#define DCONV     4
#define DTRANK    64
#define MROWS     (BB * LL)          // 2048 rows for all GEMMs

typedef float v2f __attribute__((ext_vector_type(2)));
typedef float v8f __attribute__((ext_vector_type(8)));

__device__ __forceinline__ float silu_f(float x) {
    return x / (1.0f + __expf(-x));
}
__device__ __forceinline__ float softplus_f(float x) {
    return (x > 20.0f) ? x : log1pf(__expf(x));
}

// ---------------------------------------------------------------------------
// fp32 WMMA GEMM:  C[M,N] = A[M,K] @ B[K,N] + bias[N]  (+ optional softplus)
// Template params: N, K compile-time; NT = #16-wide n-tiles per wave (A reuse);
// EPI: 0 = none, 1 = softplus.
//
// VGPR layouts (CDNA5 ISA, 16x16x4 f32):
//   A (16x4): lane m=lane&15; VGPR0 holds K = 2*(lane>>4), VGPR1 K = +1
//   B (4x16): lane n=lane&15; VGPR0 holds K = 2*(lane>>4), VGPR1 K = +1
//   C (16x16): VGPR i -> row i + 8*(lane>>4), col lane&15
// ---------------------------------------------------------------------------
template <int N, int K, int NT, int EPI>
__global__ __launch_bounds__(256)
void gemm_wmma(const float* __restrict__ A,
               const float* __restrict__ Bm,
               const float* __restrict__ bias,
               float* __restrict__ C,
               int totalWaves) {
    constexpr int NGROUPS = N / (16 * NT);
    const int wave = threadIdx.x >> 5;
    const int lane = threadIdx.x & 31;
    const int wid  = blockIdx.x * 8 + wave;
    if (wid >= totalWaves) return;                 // wave-uniform exit

    const int tm = wid / NGROUPS;                  // 16-row strip
    const int tg = wid % NGROUPS;                  // 16*NT-col strip

    const int r     = lane & 15;                   // A row / B,C col within tile
    const int khalf = lane >> 4;                   // 0: K pair {0,1}, 1: {2,3}

    const float* __restrict__ Arow  = A  + (size_t)(tm * 16 + r) * K + khalf * 2;
    const float* __restrict__ Bbase = Bm + tg * (16 * NT) + r + (size_t)(khalf * 2) * N;

    v8f acc[NT];
    #pragma unroll
    for (int t = 0; t < NT; ++t) acc[t] = (v8f){};

    #pragma unroll 4
    for (int k = 0; k < K; k += 4) {
        if ((k & 15) == 0 && k + 16 < K) {         // keep L0 warm, 1 per K-block
            __builtin_prefetch(Arow + k + 16, 0, 3);
            __builtin_prefetch(Bbase + (size_t)(k + 16) * N, 0, 3);
        }
        const v2f a = *(const v2f*)(Arow + k);     // contiguous pair -> b64 load
        #pragma unroll
        for (int t = 0; t < NT; ++t) {
            v2f b;
            b.x = Bbase[(size_t)k * N     + t * 16];
            b.y = Bbase[(size_t)k * N + N + t * 16];
            // D = A*B + C (8 args: neg_a, A, neg_b, B, c_mod, C, reuse_a, reuse_b)
            acc[t] = __builtin_amdgcn_wmma_f32_16x16x4_f32(
                false, a, false, b, (short)0, acc[t], false, false);
        }
    }

    const int crow = tm * 16 + khalf * 8;
    #pragma unroll
    for (int t = 0; t < NT; ++t) {
        const int ccol = tg * (16 * NT) + t * 16 + r;
        const float bv = bias[ccol];
        #pragma unroll
        for (int i = 0; i < 8; ++i) {
            float v = acc[t][i] + bv;
            if (EPI == 1) v = softplus_f(v);
            C[(size_t)(crow + i) * N + ccol] = v;
        }
    }
}

// ---------------------------------------------------------------------------
// Depthwise 'SAME' conv over L (kernel 4, pad_left=1, pad_right=2) + SiLU.
// x_inner is the first DINNER columns of x_proj (row stride 2*DINNER).
// conv_w layout (D_CONV, 1, D_INNER): w[k*DINNER + d].
// ---------------------------------------------------------------------------
__global__ void conv_silu_kernel(const float* __restrict__ x_proj,
                                 const float* __restrict__ cw,
                                 const float* __restrict__ cb,
                                 float* __restrict__ x_conv) {
    const int idx = blockIdx.x * blockDim.x + threadIdx.x;
    if (idx >= BB * LL * DINNER) return;
    const int d  = idx % DINNER;
    const int l  = (idx / DINNER) % LL;
    const int b  = idx / (DINNER * LL);

    float s = cb[d];
    #pragma unroll
    for (int k = 0; k < DCONV; ++k) {
        const int ls = l + k - 1;                  // SAME: pad_left = 1
        if (ls >= 0 && ls < LL)
            s += x_proj[((size_t)(b * LL + ls)) * (2 * DINNER) + d] * cw[k * DINNER + d];
    }
    x_conv[idx] = silu_f(s);
}

// ---------------------------------------------------------------------------
// Selective scan: one thread per (b, d) channel; h[NSTATE] lives in VGPRs.
// B/C row vectors for each timestep are staged in LDS (shared by the block's
// 256 channels of the same batch). Fuses the SiLU(gate) multiply.
// grid = (DINNER/256, BB), block = 256 (8 wave32 waves).
// ---------------------------------------------------------------------------
__global__ void ssm_scan_kernel(const float* __restrict__ dt,
                                const float* __restrict__ x_conv,
                                const float* __restrict__ x_proj,   // gate source
                                const float* __restrict__ Bp,
                                const float* __restrict__ Cp,
                                const float* __restrict__ A_log,
                                const float* __restrict__ D_param,
                                const float* __restrict__ ssm_state0,
                                float* __restrict__ y_gated,
                                float* __restrict__ final_state) {
    const int d = blockIdx.x * blockDim.x + threadIdx.x;
    const int b = blockIdx.y;

    float Arow[NSTATE], h[NSTATE];
    #pragma unroll
    for (int n = 0; n < NSTATE; ++n) {
        Arow[n] = -__expf(A_log[d * NSTATE + n]);
        h[n]    = ssm_state0[((size_t)(b * DINNER + d)) * NSTATE + n];
    }
    const float Dp = D_param[d];

    __shared__ float sB[NSTATE];
    __shared__ float sC[NSTATE];

    for (int l = 0; l < LL; ++l) {
        const size_t row = (size_t)(b * LL + l);
        __syncthreads();                           // protect prev-iter readers
        if (threadIdx.x < NSTATE)
            sB[threadIdx.x] = Bp[row * NSTATE + threadIdx.x];
        else if (threadIdx.x < 2 * NSTATE)
            sC[threadIdx.x - NSTATE] = Cp[row * NSTATE + (threadIdx.x - NSTATE)];
        __syncthreads();

        const float dtv = dt[row * DINNER + d];
        const float xv  = x_conv[row * DINNER + d];
        float y = 0.0f;
        #pragma unroll
        for (int n = 0; n < NSTATE; ++n) {
            const float abar = __expf(dtv * Arow[n]);
            h[n] = abar * h[n] + (dtv * sB[n]) * xv;
            y += h[n] * sC[n];
        }
        y += Dp * xv;

        const float g = x_proj[row * (2 * DINNER) + DINNER + d];
        y_gated[row * DINNER + d] = y * silu_f(g);
    }

    #pragma unroll
    for (int n = 0; n < NSTATE; ++n)
        final_state[((size_t)(b * DINNER + d)) * NSTATE + n] = h[n];
}

// ---------------------------------------------------------------------------

extern "C" void kernel_launch(void* const* d_in, const int* in_sizes, int n_in,
                              void* d_out, int out_size, void* d_ws, size_t ws_size,
                              hipStream_t stream) {
    const float* x           = (const float*)d_in[0];
    const float* ssm_state0  = (const float*)d_in[1];
    const float* in_proj_w   = (const float*)d_in[2];
    const float* in_proj_b   = (const float*)d_in[3];
    const float* conv_w      = (const float*)d_in[4];
    const float* conv_b      = (const float*)d_in[5];
    const float* dt_proj_w   = (const float*)d_in[6];
    const float* dt_proj_b   = (const float*)d_in[7];
    const float* dt_expand_w = (const float*)d_in[8];
    const float* dt_expand_b = (const float*)d_in[9];
    const float* A_log       = (const float*)d_in[10];
    const float* B_proj_w    = (const float*)d_in[11];
    const float* B_proj_b    = (const float*)d_in[12];
    const float* C_proj_w    = (const float*)d_in[13];
    const float* C_proj_b    = (const float*)d_in[14];
    const float* D_param     = (const float*)d_in[15];
    const float* out_proj_w  = (const float*)d_in[16];
    const float* out_proj_b  = (const float*)d_in[17];

    float* out         = (float*)d_out;                    // [B,L,D_MODEL]
    float* final_state = out + (size_t)BB * LL * DMODEL;   // [B,D_INNER,N]

    // workspace layout (floats), total ~21.2M floats ~= 85 MB
    float* ws      = (float*)d_ws;
    float* x_proj  = ws;                                   // 2048 x 4096
    float* x_conv  = x_proj + (size_t)MROWS * 2 * DINNER;  // 2048 x 2048
    float* dt_low  = x_conv + (size_t)MROWS * DINNER;      // 2048 x 64
    float* dt_f    = dt_low + (size_t)MROWS * DTRANK;      // 2048 x 2048
    float* Bp      = dt_f   + (size_t)MROWS * DINNER;      // 2048 x 16
    float* Cp      = Bp     + (size_t)MROWS * NSTATE;      // 2048 x 16
    float* y_g     = Cp     + (size_t)MROWS * NSTATE;      // 2048 x 2048

    const int tilesM = MROWS / 16;                         // 128

    // 1) x_proj = x @ in_proj_w + b          [2048,1024]x[1024,4096], NT=4
    {
        const int waves = tilesM * (2 * DINNER) / 64;      // 8192
        gemm_wmma<2 * DINNER, DMODEL, 4, 0>
            <<<(waves + 7) / 8, 256, 0, stream>>>(
                x, in_proj_w, in_proj_b, x_proj, waves);
    }

    // 2) depthwise conv + SiLU -> x_conv
    conv_silu_kernel<<<(BB * LL * DINNER + 255) / 256, 256, 0, stream>>>(
        x_proj, conv_w, conv_b, x_conv);

    // 3) dt_low = x_conv @ dt_proj_w + b     [2048,2048]x[2048,64], NT=4
    {
        const int waves = tilesM * DTRANK / 64;            // 128
        gemm_wmma<DTRANK, DINNER, 4, 0>
            <<<(waves + 7) / 8, 256, 0, stream>>>(
                x_conv, dt_proj_w, dt_proj_b, dt_low, waves);
    }

    // 4) dt = softplus(dt_low @ dt_expand_w + b)   [2048,64]x[64,2048], NT=4
    {
        const int waves = tilesM * DINNER / 64;            // 4096
        gemm_wmma<DINNER, DTRANK, 4, 1>
            <<<(waves + 7) / 8, 256, 0, stream>>>(
                dt_low, dt_expand_w, dt_expand_b, dt_f, waves);
    }

    // 5) Bp / Cp projections                 [2048,2048]x[2048,16], NT=1
    {
        const int waves = tilesM;                          // 128
        gemm_wmma<NSTATE, DINNER, 1, 0>
            <<<(waves + 7) / 8, 256, 0, stream>>>(
                x_conv, B_proj_w, B_proj_b, Bp, waves);
        gemm_wmma<NSTATE, DINNER, 1, 0>
            <<<(waves + 7) / 8, 256, 0, stream>>>(
                x_conv, C_proj_w, C_proj_b, Cp, waves);
    }

    // 6) sequential selective scan (fuses SiLU gate), writes final_state
    ssm_scan_kernel<<<dim3(DINNER / 256, BB), 256, 0, stream>>>(
        dt_f, x_conv, x_proj, Bp, Cp, A_log, D_param, ssm_state0,
        y_g, final_state);

    // 7) out = y_g @ out_proj_w + b          [2048,2048]x[2048,1024], NT=4
    {
        const int waves = tilesM * DMODEL / 64;            // 2048
        gemm_wmma<DMODEL, DINNER, 4, 0>
            <<<(waves + 7) / 8, 256, 0, stream>>>(
                y_g, out_proj_w, out_proj_b, out, waves);
    }
}